// CIEDE2000Loss_88244398063711
// MI455X (gfx1250) — compile-verified
//
#include <hip/hip_runtime.h>
#include <math.h>

// CIEDE2000 per-pixel loss: img1, img2 are (16, 3, 512, 512) float32, channel-planar.
// Output (16, 512, 512) float32 = deltaE/100.
// Memory-bound streaming kernel: ~117 MB traffic -> ~5us at 23.3 TB/s; working set
// fits the 192 MB L2 so replays run at L2 bandwidth. Fully branchless compute built
// on raw gfx1250 hardware transcendentals (v_exp_f32 / v_log_f32 / v_sin_f32 /
// v_cos_f32 / v_sqrt_f32 / v_rcp_f32) so VALU work stays under the memory roofline.

#define PI_F      3.14159265358979323846f
#define TWO_PI_F  6.28318530717958647692f
#define HALF_PI_F 1.57079632679489661923f
#define INV_2PI_F 0.15915494309189535f
#define LOG2E_F   1.4426950408889634f
#define POW25_7   6103515625.0f            // 25^7

// ---- raw hardware op wrappers (single v_* instruction each) ----
__device__ __forceinline__ float hw_rcp (float x) { return __builtin_amdgcn_rcpf(x); }
__device__ __forceinline__ float hw_sqrt(float x) { return __builtin_amdgcn_sqrtf(x); }
__device__ __forceinline__ float hw_exp2(float x) { return __builtin_amdgcn_exp2f(x); }
__device__ __forceinline__ float hw_log2(float x) { return __builtin_amdgcn_logf(x); }
__device__ __forceinline__ float hw_sin (float x) { return __builtin_amdgcn_sinf(x * INV_2PI_F); }
__device__ __forceinline__ float hw_cos (float x) { return __builtin_amdgcn_cosf(x * INV_2PI_F); }
__device__ __forceinline__ float fast_div(float a, float b) { return a * hw_rcp(b); }
// x > 0 required; our args are in [1e-8, ~1.1] -> no denorm/overflow edge cases
__device__ __forceinline__ float fast_pow(float x, float k) { return hw_exp2(k * hw_log2(x)); }

// Branchless atan2 via minimax polynomial on [0,1]; atan2(0,0) = 0 like jnp.
__device__ __forceinline__ float fast_atan2(float y, float x) {
    float ax = fabsf(x), ay = fabsf(y);
    float mx = fmaxf(ax, ay), mn = fminf(ax, ay);
    float r  = mn * hw_rcp(mx);            // 0/0 -> NaN, patched below
    float s  = r * r;
    float p  =            -0.01172120f;
    p = p * s + 0.05265332f;
    p = p * s - 0.11643287f;
    p = p * s + 0.19354346f;
    p = p * s - 0.33262347f;
    p = p * s + 0.99997726f;
    float th = p * r;                       // atan(mn/mx) in [0, pi/4]
    th = (ay > ax)    ? (HALF_PI_F - th) : th;
    th = (x < 0.0f)   ? (PI_F - th)      : th;
    th = (mx == 0.0f) ? 0.0f             : th;
    return (y < 0.0f) ? -th : th;
}

// sRGB -> linear, branchless (pow path taken by ~96% of uniform lanes anyway)
__device__ __forceinline__ float srgb_lin(float u) {
    float p = fast_pow((u + 0.055f) * (1.0f / 1.055f), 2.4f);
    float l = u * (1.0f / 12.92f);
    return (u > 0.04045f) ? p : l;
}

// Lab f(t), branchless
__device__ __forceinline__ float f_lab(float t) {
    float c = fast_pow(fmaxf(t, 1e-8f), 1.0f / 3.0f);
    float l = 7.787f * t + (4.0f / 29.0f);
    return (t > 0.008856f) ? c : l;
}

__device__ __forceinline__ void rgb_to_lab(float r, float g, float b,
                                           float& L, float& a, float& bb) {
    r = srgb_lin(r); g = srgb_lin(g); b = srgb_lin(b);
    float x = 0.412453f * r + 0.35758f  * g + 0.180423f * b;
    float y = 0.212671f * r + 0.71516f  * g + 0.072169f * b;
    float z = 0.019334f * r + 0.119193f * g + 0.950227f * b;
    float fx = f_lab(x * (1.0f / 0.95047f));
    float fy = f_lab(y);
    float fz = f_lab(z * (1.0f / 1.08883f));
    L  = 116.0f * fy - 16.0f;
    a  = 500.0f * (fx - fy);
    bb = 200.0f * (fy - fz);
}

__device__ __forceinline__ float pow7(float x) {
    float x2 = x * x;
    float x4 = x2 * x2;
    return x4 * x2 * x;
}

__device__ __forceinline__ float ciede_pixel(float r1, float g1, float b1c,
                                             float r2, float g2, float b2c) {
    float L1, a1, b1, L2, a2, b2;
    rgb_to_lab(r1, g1, b1c, L1, a1, b1);
    rgb_to_lab(r2, g2, b2c, L2, a2, b2);

    float C1   = hw_sqrt(a1 * a1 + b1 * b1);
    float C2   = hw_sqrt(a2 * a2 + b2 * b2);
    float Cbar = 0.5f * (C1 + C2);
    float c7   = pow7(Cbar);
    float G    = 0.5f * (1.0f - hw_sqrt(fast_div(c7, c7 + POW25_7)));
    float scale = 1.0f + G;
    float a1p  = a1 * scale;
    float a2p  = a2 * scale;
    float C1p  = hw_sqrt(a1p * a1p + b1 * b1);
    float C2p  = hw_sqrt(a2p * a2p + b2 * b2);

    float h1 = fast_atan2(b1, a1p); h1 = (h1 < 0.0f) ? h1 + TWO_PI_F : h1;
    float h2 = fast_atan2(b2, a2p); h2 = (h2 < 0.0f) ? h2 + TWO_PI_F : h2;

    float Lbar   = 0.5f * (L1 + L2);
    float tmp    = (Lbar - 50.0f) * (Lbar - 50.0f);
    float SL     = 1.0f + fast_div(0.015f * tmp, hw_sqrt(20.0f + tmp));
    float L_term = fast_div(L2 - L1, SL);

    float Cbarp  = 0.5f * (C1p + C2p);
    float SC     = 1.0f + 0.045f * Cbarp;
    float C_term = fast_div(C2p - C1p, SC);

    float h_diff = h2 - h1;
    float h_sum  = h1 + h2;
    float CC     = C1p * C2p;

    float dH = h_diff;
    dH = (h_diff >  PI_F) ? h_diff - TWO_PI_F : dH;
    dH = (h_diff < -PI_F) ? h_diff + TWO_PI_F : dH;
    dH = (CC == 0.0f)     ? 0.0f              : dH;
    float dH_term = 2.0f * hw_sqrt(CC) * hw_sin(0.5f * dH);

    bool  mask   = (CC != 0.0f) && (fabsf(h_diff) > PI_F);
    float wrapped = (h_sum < TWO_PI_F) ? (h_sum + TWO_PI_F) : (h_sum - TWO_PI_F);
    float Hbar   = mask ? wrapped : h_sum;
    Hbar = (CC == 0.0f) ? h_sum * 2.0f : Hbar;
    Hbar *= 0.5f;

    float T = 1.0f
            - 0.17f * hw_cos(Hbar - PI_F / 6.0f)
            + 0.24f * hw_cos(2.0f * Hbar)
            + 0.32f * hw_cos(3.0f * Hbar + PI_F / 30.0f)
            - 0.20f * hw_cos(4.0f * Hbar - 63.0f * PI_F / 180.0f);
    float SH     = 1.0f + 0.015f * Cbarp * T;
    float H_term = fast_div(dH_term, SH);

    float c7p = pow7(Cbarp);
    float Rc  = 2.0f * hw_sqrt(fast_div(c7p, c7p + POW25_7));
    float e   = (Hbar * (180.0f / PI_F) - 275.0f) * (1.0f / 25.0f);
    float dtheta = (PI_F / 6.0f) * hw_exp2(-(e * e) * LOG2E_F);
    float R_term = -hw_sin(2.0f * dtheta) * Rc * C_term * H_term;

    float dE2 = L_term * L_term + C_term * C_term + H_term * H_term + R_term;
    return hw_sqrt(fmaxf(dE2, 0.0f)) * 0.01f;
}

// Plane = 512*512 floats = 65536 float4s per (batch, channel).
// Each thread handles one float4 = 4 pixels: 6x global_load_b128 + 1x global_store_b128.
__global__ void __launch_bounds__(256)
ciede2000_kernel(const float4* __restrict__ img1,
                 const float4* __restrict__ img2,
                 float4* __restrict__ out,
                 int total_quads) {
    int q = blockIdx.x * blockDim.x + threadIdx.x;
    if (q >= total_quads) return;

    const int PLANE_Q = 65536;            // 512*512/4
    int batch  = q >> 16;
    int within = q & 0xFFFF;
    int base   = batch * (3 * PLANE_Q) + within;

    float4 r1 = img1[base];
    float4 g1 = img1[base + PLANE_Q];
    float4 b1 = img1[base + 2 * PLANE_Q];
    float4 r2 = img2[base];
    float4 g2 = img2[base + PLANE_Q];
    float4 b2 = img2[base + 2 * PLANE_Q];

    float4 o;
    o.x = ciede_pixel(r1.x, g1.x, b1.x, r2.x, g2.x, b2.x);
    o.y = ciede_pixel(r1.y, g1.y, b1.y, r2.y, g2.y, b2.y);
    o.z = ciede_pixel(r1.z, g1.z, b1.z, r2.z, g2.z, b2.z);
    o.w = ciede_pixel(r1.w, g1.w, b1.w, r2.w, g2.w, b2.w);

    out[q] = o;
}

extern "C" void kernel_launch(void* const* d_in, const int* in_sizes, int n_in,
                              void* d_out, int out_size, void* d_ws, size_t ws_size,
                              hipStream_t stream) {
    (void)n_in; (void)d_ws; (void)ws_size; (void)in_sizes;
    const float4* img1 = (const float4*)d_in[0];
    const float4* img2 = (const float4*)d_in[1];
    float4* out = (float4*)d_out;

    int total_quads = out_size / 4;       // 4,194,304 pixels -> 1,048,576 quads
    int threads = 256;                    // 8 wave32s per block
    int blocks  = (total_quads + threads - 1) / threads;
    ciede2000_kernel<<<blocks, threads, 0, stream>>>(img1, img2, out, total_quads);
}